// LightGCN_xij_item_personal_matrix_50294067036543
// MI455X (gfx1250) — compile-verified
//
#include <hip/hip_runtime.h>
#include <cstdint>
#include <cstddef>

// ---------------------------------------------------------------------------
// LightGCN-style pipeline for MI455X (gfx1250, wave32).
//   e0 = concat(emb_user, emb_item)                [N,64]
//   e{k+1} = SpMM(G, e{k})   (atomic f32 scatter, L2-resident vectors)
//   head: gather 0.25*(e0+e1+e2+e3) rows, fp32 WMMA 16x16x4 GEMMs vs
//         w_user^T / w_item^T, softmax/sigmoid epilogue -> gamma[B]
// ---------------------------------------------------------------------------

typedef __attribute__((ext_vector_type(2))) float v2f;
typedef __attribute__((ext_vector_type(8))) float v8f;

constexpr int DIM    = 64;
constexpr int LSTR   = 68;   // LDS row stride (floats): 16B-aligned, bank-spread

// ---------------- init: e0 = concat(emb_user, emb_item) --------------------
__global__ __launch_bounds__(256)
void k_init_concat(const float4* __restrict__ eu, const float4* __restrict__ ei,
                   float4* __restrict__ e0, long long n4u, long long n4tot) {
    long long i = (long long)blockIdx.x * blockDim.x + threadIdx.x;
    if (i >= n4tot) return;
    e0[i] = (i < n4u) ? eu[i] : ei[i - n4u];
}

// ---------------- zero a buffer --------------------------------------------
__global__ __launch_bounds__(256)
void k_zero(float4* __restrict__ p, long long n4) {
    long long i = (long long)blockIdx.x * blockDim.x + threadIdx.x;
    if (i >= n4) return;
    p[i] = make_float4(0.f, 0.f, 0.f, 0.f);
}

// ---------------- SpMM scatter: out[row] += val * in[col] ------------------
// one wave32 per edge, float2 per lane; f32 atomics execute in L2
__global__ __launch_bounds__(256)
void k_spmm(const float* __restrict__ vals, const int* __restrict__ rows,
            const int* __restrict__ cols, const float* __restrict__ xin,
            float* __restrict__ xout, int E) {
    int e = blockIdx.x * 8 + (threadIdx.x >> 5);
    if (e >= E) return;
    int lane = threadIdx.x & 31;
    float v = vals[e];
    int r = rows[e];
    int c = cols[e];
    const float2 x = *(const float2*)(xin + (size_t)c * DIM + lane * 2);
    float* o = xout + (size_t)r * DIM + lane * 2;
    unsafeAtomicAdd(o,     v * x.x);   // global_atomic_add_f32
    unsafeAtomicAdd(o + 1, v * x.y);
}

// ---------------- fused head: gather + WMMA GEMMs + epilogue ---------------
// 128 threads (4 waves) per block; each block handles 16 batch rows.
__global__ __launch_bounds__(128)
void k_head(const float* __restrict__ e0, const float* __restrict__ e1,
            const float* __restrict__ e2, const float* __restrict__ e3,
            const float* __restrict__ w_user, const float* __restrict__ w_item,
            const float* __restrict__ xij0, const float* __restrict__ xij1,
            const int* __restrict__ users, const int* __restrict__ items,
            const int* __restrict__ xij, float* __restrict__ out,
            int U, int B) {
    __shared__ float smem[4 * 16 * LSTR];
    float* sAu = smem;                 // gathered user rows   [16][LSTR]
    float* sAi = smem + 16 * LSTR;     // gathered item rows
    float* sOu = smem + 32 * LSTR;     // users_emb @ w_user^T
    float* sOi = smem + 48 * LSTR;     // items_emb @ w_item^T

    const int tid = threadIdx.x;
    const int b0  = blockIdx.x * 16;

    // ---- stage 1: gather light_out rows = 0.25*(e0+e1+e2+e3) ----
    {
        int r  = tid >> 3;            // 0..15  (batch row within block)
        int cs = (tid & 7) * 8;       // 8 floats per thread
        int b  = b0 + r;
        if (b < B) {
            size_t ru = (size_t)users[b] * DIM;
            size_t ri = (size_t)(U + items[b]) * DIM;
            #pragma unroll
            for (int q = 0; q < 8; q += 4) {
                float4 a = *(const float4*)(e0 + ru + cs + q);
                float4 bv = *(const float4*)(e1 + ru + cs + q);
                float4 cv = *(const float4*)(e2 + ru + cs + q);
                float4 dv = *(const float4*)(e3 + ru + cs + q);
                float4 s;
                s.x = 0.25f * (a.x + bv.x + cv.x + dv.x);
                s.y = 0.25f * (a.y + bv.y + cv.y + dv.y);
                s.z = 0.25f * (a.z + bv.z + cv.z + dv.z);
                s.w = 0.25f * (a.w + bv.w + cv.w + dv.w);
                *(float4*)(sAu + r * LSTR + cs + q) = s;

                a  = *(const float4*)(e0 + ri + cs + q);
                bv = *(const float4*)(e1 + ri + cs + q);
                cv = *(const float4*)(e2 + ri + cs + q);
                dv = *(const float4*)(e3 + ri + cs + q);
                s.x = 0.25f * (a.x + bv.x + cv.x + dv.x);
                s.y = 0.25f * (a.y + bv.y + cv.y + dv.y);
                s.z = 0.25f * (a.z + bv.z + cv.z + dv.z);
                s.w = 0.25f * (a.w + bv.w + cv.w + dv.w);
                *(float4*)(sAi + r * LSTR + cs + q) = s;
            }
        } else {
            #pragma unroll
            for (int q = 0; q < 8; q += 4) {
                *(float4*)(sAu + r * LSTR + cs + q) = make_float4(0.f, 0.f, 0.f, 0.f);
                *(float4*)(sAi + r * LSTR + cs + q) = make_float4(0.f, 0.f, 0.f, 0.f);
            }
        }
    }
    __syncthreads();

    // ---- stage 2: fp32 WMMA 16x16x4, K=64 -> 16 steps; two chains ----
    {
        const int lane = tid & 31;
        const int wv   = tid >> 5;         // 0..3 -> output column block
        const int nb   = wv * 16;
        const int l    = lane & 15;
        const int koff = (lane >> 4) << 1; // 0 (lanes 0-15) or 2 (lanes 16-31)

        v8f cu = {};
        v8f ci = {};
        #pragma unroll
        for (int k0 = 0; k0 < DIM; k0 += 4) {
            int kb = k0 + koff;
            // A frag: (M=l, K=kb..kb+1) from LDS; B frag: out = x @ W^T, so
            // B[k][n] = W[(nb+n)*64 + k] -> contiguous float2 at row nb+l.
            v2f au = *(const v2f*)(sAu + l * LSTR + kb);
            v2f bu = *(const v2f*)(w_user + (size_t)(nb + l) * DIM + kb);
            cu = __builtin_amdgcn_wmma_f32_16x16x4_f32(
                     false, au, false, bu, (short)0, cu, false, false);
            v2f ai = *(const v2f*)(sAi + l * LSTR + kb);
            v2f bi = *(const v2f*)(w_item + (size_t)(nb + l) * DIM + kb);
            ci = __builtin_amdgcn_wmma_f32_16x16x4_f32(
                     false, ai, false, bi, (short)0, ci, false, false);
        }
        // C/D layout: VGPR r -> M = r + 8*(lane>=16), N = lane&15
        const int mbase = (lane >> 4) << 3;
        #pragma unroll
        for (int r = 0; r < 8; ++r) {
            sOu[(mbase + r) * LSTR + nb + l] = cu[r];
            sOi[(mbase + r) * LSTR + nb + l] = ci[r];
        }
    }
    __syncthreads();

    // ---- stage 3: softmax / sigmoid / gamma ----
    if (tid < 16) {
        int b = b0 + tid;
        if (b < B) {
            const float* ru = sOu + tid * LSTR;
            const float* ri = sOi + tid * LSTR;
            float mx = -3.402823466e+38f;
            #pragma unroll 4
            for (int j = 0; j < DIM; ++j) mx = fmaxf(mx, ru[j]);
            float sum = 0.f;
            #pragma unroll 4
            for (int j = 0; j < DIM; ++j) sum = __fmaf_rn(1.f, expf(ru[j] - mx), sum);
            float inv = 0.5f / sum;                 // (1-hx)=0.5 folded in
            float g = 0.f;
            #pragma unroll 4
            for (int j = 0; j < DIM; ++j) {
                float u  = expf(ru[j] - mx) * inv;
                float it = 1.f / (1.f + expf(-ri[j]));
                g = __fmaf_rn(u, it, g);
            }
            int   item = items[b];
            float xe   = xij[b] ? xij1[item] : xij0[item];
            g += 0.5f * (1.f / (1.f + expf(-xe)));  // hx * sigmoid(last col)
            out[b] = g;
        }
    }
}

// ---------------------------------------------------------------------------
extern "C" void kernel_launch(void* const* d_in, const int* in_sizes, int n_in,
                              void* d_out, int out_size, void* d_ws, size_t ws_size,
                              hipStream_t stream) {
    const float* emb_user = (const float*)d_in[0];
    const float* emb_item = (const float*)d_in[1];
    const float* xij0     = (const float*)d_in[2];
    const float* xij1     = (const float*)d_in[3];
    const float* w_user   = (const float*)d_in[4];
    const float* w_item   = (const float*)d_in[5];
    const float* edge_v   = (const float*)d_in[6];
    const int*   edge_r   = (const int*)d_in[7];
    const int*   edge_c   = (const int*)d_in[8];
    const int*   users    = (const int*)d_in[9];
    const int*   items    = (const int*)d_in[10];
    const int*   xij      = (const int*)d_in[11];
    float*       out      = (float*)d_out;

    const int U = in_sizes[0] / DIM;
    const int I = in_sizes[1] / DIM;
    const int E = in_sizes[6];
    const int B = in_sizes[9];
    const long long N  = (long long)U + I;
    const long long ND = N * DIM;          // floats per layer buffer
    const long long N4 = ND / 4;

    float* e0 = (float*)d_ws;
    float* e1 = e0 + ND;
    float* e2 = e1 + ND;
    float* e3 = e2 + ND;

    // e0 = concat(emb_user, emb_item)
    {
        long long n4u = (long long)U * DIM / 4;
        long long blocks = (N4 + 255) / 256;
        k_init_concat<<<(unsigned)blocks, 256, 0, stream>>>(
            (const float4*)emb_user, (const float4*)emb_item, (float4*)e0, n4u, N4);
    }

    const unsigned zb = (unsigned)((N4 + 255) / 256);
    const unsigned sb = (unsigned)((E + 7) / 8);
    float* bufs[4] = {e0, e1, e2, e3};
    for (int layer = 0; layer < 3; ++layer) {
        k_zero<<<zb, 256, 0, stream>>>((float4*)bufs[layer + 1], N4);
        k_spmm<<<sb, 256, 0, stream>>>(edge_v, edge_r, edge_c,
                                       bufs[layer], bufs[layer + 1], E);
    }

    k_head<<<(unsigned)((B + 15) / 16), 128, 0, stream>>>(
        e0, e1, e2, e3, w_user, w_item, xij0, xij1,
        users, items, xij, out, U, B);
}